// TorchRNN_19207093748374
// MI455X (gfx1250) — compile-verified
//
#include <hip/hip_runtime.h>
#include <cmath>

typedef __attribute__((ext_vector_type(2))) float v2f;
typedef __attribute__((ext_vector_type(8))) float v8f;

#define VOCAB   50257
#define EMBED   128
#define HIDDEN  128
#define NCLS    4
#define BB      256
#define TT      512
#define LDSROW  132   // stride ≡ 4 mod 64 banks -> 16-row b64 A-fragment reads hit 64 distinct banks

// ---------------------------------------------------------------------------
// Kernel 1: embedding gather fused with input projection.
//   x_proj[t, b, n] = sum_k emb[x[b,t]][k] * W_ih[n][k] + b_ih[n] + b_hh[n]
// NOTE: output is written TIME-MAJOR [T, B, H] so the serial recurrence reads
// one contiguous 8 KB tile per step. One block = one 16-row M-tile of the
// flattened [B*T] dimension; wave w handles N-tile n0 = 16*w (8 waves).
// ---------------------------------------------------------------------------
__global__ void __launch_bounds__(256) embed_proj_kernel(
    const int* __restrict__ x, const float* __restrict__ emb,
    const float* __restrict__ W_ih, const float* __restrict__ b_ih,
    const float* __restrict__ b_hh, float* __restrict__ x_proj)
{
  const int wave  = threadIdx.x >> 5;
  const int lane  = threadIdx.x & 31;
  const int m0    = blockIdx.x * 16;
  const int n0    = wave * 16;
  const int col   = lane & 15;
  const int khalf = lane >> 4;

  const int tok     = x[m0 + col];                     // A row for this lane
  const float* arow = emb  + (size_t)tok * EMBED;
  const int    n    = n0 + col;
  const float* brow = W_ih + (size_t)n * EMBED;        // B[k][n] = W_ih[n][k]
  const float  bias = b_ih[n] + b_hh[n];

  v8f acc = {};
#pragma unroll
  for (int k0 = 0; k0 < EMBED; k0 += 4) {
    v2f a = *(const v2f*)(arow + k0 + 2 * khalf);
    v2f b = *(const v2f*)(brow + k0 + 2 * khalf);
    acc = __builtin_amdgcn_wmma_f32_16x16x4_f32(false, a, false, b,
                                                (short)0, acc, false, false);
  }
#pragma unroll
  for (int v = 0; v < 8; ++v) {
    int m = m0 + v + 8 * khalf;                        // flattened b*T + t
    int b = m >> 9;                                    // T = 512
    int t = m & (TT - 1);
    x_proj[((size_t)t * BB + b) * HIDDEN + n] = acc[v] + bias;  // [T,B,H]
  }
}

// ---------------------------------------------------------------------------
// Kernel 2: tanh-RNN recurrence, batch-parallel across blocks.
// Block owns 16 batch rows; h double-buffered in LDS; W_hh^T fragments live
// in 64 VGPRs for all 512 steps. xp tile register-prefetched one step ahead,
// cache-prefetched two steps ahead. One workgroup barrier per step.
// ---------------------------------------------------------------------------
__global__ void __launch_bounds__(256) rnn_kernel(
    const float* __restrict__ h0, const float* __restrict__ W_hh,
    const float* __restrict__ x_proj, float* __restrict__ last_hidden)
{
  __shared__ float hsh[2][16][LDSROW];

  const int lane  = threadIdx.x & 31;
  const int wave  = threadIdx.x >> 5;
  const int b0    = blockIdx.x * 16;
  const int n0    = wave * 16;
  const int col   = lane & 15;
  const int khalf = lane >> 4;
  const int n     = n0 + col;

  // h0 -> LDS buffer 0
  for (int i = threadIdx.x; i < 16 * HIDDEN; i += 256) {
    int r = i >> 7, c = i & (HIDDEN - 1);
    hsh[0][r][c] = h0[(size_t)(b0 + r) * HIDDEN + c];
  }

  // Resident B fragments: B[k][n] = W_hh[n][k]
  v2f breg[32];
  const float* brow = W_hh + (size_t)n * HIDDEN;
#pragma unroll
  for (int kk = 0; kk < 32; ++kk)
    breg[kk] = *(const v2f*)(brow + kk * 4 + 2 * khalf);

  // Per-lane base into time-major x_proj [T, B, H]; step stride = B*H.
  const float* xp_base = x_proj + (size_t)(b0 + 8 * khalf) * HIDDEN + n;
  const size_t tstride = (size_t)BB * HIDDEN;

  float xp_next[8];
#pragma unroll
  for (int v = 0; v < 8; ++v) xp_next[v] = xp_base[(size_t)v * HIDDEN];  // t = 0

  __syncthreads();

  int cur = 0;
  for (int t = 0; t < TT; ++t) {
    // consume this step's prefetch; immediately issue next step's loads
    float xp_cur[8];
#pragma unroll
    for (int v = 0; v < 8; ++v) xp_cur[v] = xp_next[v];
    if (t + 1 < TT) {
      const float* bp = xp_base + (size_t)(t + 1) * tstride;
#pragma unroll
      for (int v = 0; v < 8; ++v) xp_next[v] = bp[(size_t)v * HIDDEN];
    }
    if (t + 2 < TT)
      __builtin_prefetch(xp_base + (size_t)(t + 2) * tstride, 0, 1);

    v8f acc = {};
#pragma unroll
    for (int kk = 0; kk < 32; ++kk) {
      v2f a = *(const v2f*)(&hsh[cur][col][kk * 4 + 2 * khalf]);
      acc = __builtin_amdgcn_wmma_f32_16x16x4_f32(false, a, false, breg[kk],
                                                  (short)0, acc, false, false);
    }

    const int nxt = cur ^ 1;
#pragma unroll
    for (int v = 0; v < 8; ++v) {
      int m = v + 8 * khalf;
      float hn = tanhf(acc[v] + xp_cur[v]);
      hsh[nxt][m][n] = hn;
      if (t == TT - 1)
        last_hidden[(size_t)(b0 + m) * HIDDEN + n] = hn;
    }
    __syncthreads();
    cur = nxt;
  }
}

// ---------------------------------------------------------------------------
// Kernel 3a: hidden_mlp = relu(last_hidden @ W1^T + b1)   [256 x 256]
// ---------------------------------------------------------------------------
__global__ void __launch_bounds__(256) mlp1_kernel(
    const float* __restrict__ lh, const float* __restrict__ W1,
    const float* __restrict__ b1, float* __restrict__ hm)
{
  __shared__ float row[HIDDEN];
  const int b = blockIdx.x;
  const int j = threadIdx.x;
  if (j < HIDDEN) row[j] = lh[(size_t)b * HIDDEN + j];
  __syncthreads();
  float s = b1[j];
  const float* w = W1 + (size_t)j * HIDDEN;
#pragma unroll 8
  for (int k = 0; k < HIDDEN; ++k) s = fmaf(row[k], w[k], s);
  hm[(size_t)b * (2 * HIDDEN) + j] = fmaxf(s, 0.0f);
}

// ---------------------------------------------------------------------------
// Kernel 3b: logits = hidden_mlp @ W2^T + b2   [256 x 4]
// ---------------------------------------------------------------------------
__global__ void __launch_bounds__(256) mlp2_kernel(
    const float* __restrict__ hm, const float* __restrict__ W2,
    const float* __restrict__ b2, float* __restrict__ logits)
{
  const int tid = blockIdx.x * 256 + threadIdx.x;      // 1024 = 256 * 4
  const int b = tid >> 2, c = tid & 3;
  float s = b2[c];
  const float* h = hm + (size_t)b * (2 * HIDDEN);
  const float* w = W2 + (size_t)c * (2 * HIDDEN);
#pragma unroll 8
  for (int k = 0; k < 2 * HIDDEN; ++k) s = fmaf(h[k], w[k], s);
  logits[(size_t)b * NCLS + c] = s;
}

// ---------------------------------------------------------------------------
extern "C" void kernel_launch(void* const* d_in, const int* in_sizes, int n_in,
                              void* d_out, int out_size, void* d_ws, size_t ws_size,
                              hipStream_t stream) {
  (void)in_sizes; (void)n_in; (void)out_size; (void)ws_size;
  const int*   x    = (const int*)  d_in[0];
  const float* h0   = (const float*)d_in[1];
  const float* emb  = (const float*)d_in[2];
  const float* W_ih = (const float*)d_in[3];
  const float* W_hh = (const float*)d_in[4];
  const float* b_ih = (const float*)d_in[5];
  const float* b_hh = (const float*)d_in[6];
  const float* W1   = (const float*)d_in[7];
  const float* b1   = (const float*)d_in[8];
  const float* W2   = (const float*)d_in[9];
  const float* b2   = (const float*)d_in[10];

  float* out         = (float*)d_out;
  float* logits      = out;                       // [256, 4]
  float* last_hidden = out + BB * NCLS;           // [256, 128]

  float* x_proj = (float*)d_ws;                               // [T,B,H] f32 = 64 MB
  float* hm     = x_proj + (size_t)BB * TT * HIDDEN;          // [B,2H]  f32 = 256 KB

  embed_proj_kernel<<<(BB * TT) / 16, 256, 0, stream>>>(x, emb, W_ih, b_ih, b_hh, x_proj);
  rnn_kernel<<<BB / 16, 256, 0, stream>>>(h0, W_hh, x_proj, last_hidden);
  mlp1_kernel<<<BB, 256, 0, stream>>>(last_hidden, W1, b1, hm);
  mlp2_kernel<<<4, 256, 0, stream>>>(hm, W2, b2, logits);
}